// Brain_3624952398012
// MI455X (gfx1250) — compile-verified
//
#include <hip/hip_runtime.h>
#include <math.h>

typedef unsigned int v4u __attribute__((ext_vector_type(4)));
typedef int          v8i __attribute__((ext_vector_type(8)));
typedef int          v4i __attribute__((ext_vector_type(4)));

// ---------------------------------------------------------------------------
// K0: zero the B*N accumulator (workspace) and the B*OUT output buffer.
// Regular (RT) stores: we WANT total1 resident in L2 for step 2's gathers.
// ---------------------------------------------------------------------------
__global__ void k_zero(float* __restrict__ total1,
                       const int* __restrict__ pN, int BIN,
                       const int* __restrict__ pIN,
                       float* __restrict__ dout, int out_elems)
{
    const int N  = *pN;
    const int IN = *pIN;
    const int B  = BIN / IN;
    const long long tot = (long long)B * (long long)N;
    const long long n4  = tot >> 2;
    const long long tid    = (long long)blockIdx.x * blockDim.x + threadIdx.x;
    const long long stride = (long long)gridDim.x * blockDim.x;

    float4* t4 = (float4*)total1;
    const float4 z4 = make_float4(0.f, 0.f, 0.f, 0.f);
    for (long long i = tid; i < n4; i += stride) t4[i] = z4;
    for (long long i = (n4 << 2) + tid; i < tot; i += stride) total1[i] = 0.f;
    for (long long i = tid; i < out_elems; i += stride) dout[i] = 0.f;
}

// ---------------------------------------------------------------------------
// K1: step 1.  Stage the B*IN input matrix into LDS with the Tensor Data
// Mover (one tensor_load_to_lds per block, wave 0 issues it), then stream
// from[] (non-temporal) and scatter w*input into total1 for the rare edges
// with from < IN (~0.1% of E for uniform random indices).
// ---------------------------------------------------------------------------
__global__ void k_step1(const float* __restrict__ input,
                        const float* __restrict__ w,
                        const int*   __restrict__ from,
                        const int*   __restrict__ to,
                        float*       __restrict__ total1,
                        int E, int BIN,
                        const int* __restrict__ pN,
                        const int* __restrict__ pIN)
{
    extern __shared__ float s_in[];
    const int N  = *pN;
    const int IN = *pIN;
    const int B  = BIN / IN;

    if (BIN <= 0xFFFF) {
        // --- TDM path: one DMA of BIN floats, global -> LDS ---------------
        if (threadIdx.x < (unsigned)warpSize) {
            unsigned           lds = (unsigned)(unsigned long long)(void*)s_in; // LDS byte offset
            unsigned long long ga  = (unsigned long long)input;

            // D# group 0: count=1(valid), lds_addr, global_addr[56:0], type=2
            v4u g0;
            g0[0] = 1u;                                   // count=1, user mode
            g0[1] = lds;                                  // lds_addr
            g0[2] = (unsigned)(ga & 0xFFFFFFFFull);       // global_addr[31:0]
            g0[3] = (unsigned)((ga >> 32) & 0x01FFFFFFull) | (2u << 30); // addr[56:32] | type=2

            // D# group 1: data_size=4B, tensor BINx1, tile BINx1, stride0=BIN
            v8i g1;
            g1[0] = 0x20000;                              // data_size=2 (4 bytes)
            g1[1] = (int)((unsigned)(BIN & 0xFFFF) << 16);    // tensor_dim0[15:0] @ [63:48]
            g1[2] = (int)(((unsigned)BIN >> 16) | (1u << 16)); // tensor_dim0[31:16] | tensor_dim1=1
            g1[3] = (int)((unsigned)(BIN & 0xFFFF) << 16);    // tile_dim0 @ [127:112]
            g1[4] = 1;                                    // tile_dim1=1, tile_dim2=0
            g1[5] = BIN;                                  // tensor_dim0_stride[31:0]
            g1[6] = 0;
            g1[7] = 0;

            v4i gz4 = {0, 0, 0, 0};
            v8i gz8 = {0, 0, 0, 0, 0, 0, 0, 0};
            __builtin_amdgcn_tensor_load_to_lds(g0, g1, gz4, gz4, gz8, 0);
            __builtin_amdgcn_s_wait_tensorcnt(0);
        }
    } else {
        // --- generic fallback: cooperative LDS fill -----------------------
        for (int i = threadIdx.x; i < BIN; i += blockDim.x) s_in[i] = input[i];
    }
    __syncthreads();

    const long long tid    = (long long)blockIdx.x * blockDim.x + threadIdx.x;
    const long long stride = (long long)gridDim.x * blockDim.x;
    const long long E4 = E >> 2;
    const v4i* from4 = (const v4i*)from;

    for (long long i = tid; i < E4; i += stride) {
        const v4i f4 = __builtin_nontemporal_load(from4 + i);   // TH_NT stream
        const long long e = i << 2;
        #pragma unroll
        for (int k = 0; k < 4; ++k) {
            const int f = f4[k];
            if (f < IN) {                       // ~0.1% of edges
                const float wt = w[e + k];
                const int   t  = to[e + k];
                for (int b = 0; b < B; ++b)
                    atomicAdd(&total1[(long long)b * N + t], wt * s_in[b * IN + f]);
            }
        }
    }
    // tail
    for (long long e = (E4 << 2) + tid; e < E; e += stride) {
        const int f = from[e];
        if (f < IN) {
            const float wt = w[e];
            const int   t  = to[e];
            for (int b = 0; b < B; ++b)
                atomicAdd(&total1[(long long)b * N + t], wt * s_in[b * IN + f]);
        }
    }
}

// ---------------------------------------------------------------------------
// K2: step 2.  Stream to[] (non-temporal); for the rare edges with
// to >= N-OUT, gather tanh(total1[b*N+from]) (L2 resident) and atomically
// accumulate into d_out.
// ---------------------------------------------------------------------------
__global__ void k_step2(const float* __restrict__ w,
                        const int*   __restrict__ from,
                        const int*   __restrict__ to,
                        const float* __restrict__ total1,
                        float*       __restrict__ dout,
                        int E, int BIN,
                        const int* __restrict__ pN,
                        const int* __restrict__ pIN,
                        const int* __restrict__ pOUT)
{
    const int N   = *pN;
    const int IN  = *pIN;
    const int OUT = *pOUT;
    const int B   = BIN / IN;
    const int base = N - OUT;

    const long long tid    = (long long)blockIdx.x * blockDim.x + threadIdx.x;
    const long long stride = (long long)gridDim.x * blockDim.x;
    const long long E4 = E >> 2;
    const v4i* to4 = (const v4i*)to;

    for (long long i = tid; i < E4; i += stride) {
        const v4i t4 = __builtin_nontemporal_load(to4 + i);     // TH_NT stream
        const long long e = i << 2;
        #pragma unroll
        for (int k = 0; k < 4; ++k) {
            const int t = t4[k];
            if (t >= base) {                    // ~0.1% of edges
                const float wt = w[e + k];
                const int   f  = from[e + k];
                const int   j  = t - base;
                for (int b = 0; b < B; ++b) {
                    const float a = tanhf(total1[(long long)b * N + f]);
                    atomicAdd(&dout[b * OUT + j], wt * a);
                }
            }
        }
    }
    // tail
    for (long long e = (E4 << 2) + tid; e < E; e += stride) {
        const int t = to[e];
        if (t >= base) {
            const float wt = w[e];
            const int   f  = from[e];
            const int   j  = t - base;
            for (int b = 0; b < B; ++b) {
                const float a = tanhf(total1[(long long)b * N + f]);
                atomicAdd(&dout[b * OUT + j], wt * a);
            }
        }
    }
}

// ---------------------------------------------------------------------------
// K3: final tanh over the B*OUT outputs.
// ---------------------------------------------------------------------------
__global__ void k_final(float* __restrict__ dout, int n)
{
    const int i = blockIdx.x * blockDim.x + threadIdx.x;
    if (i < n) dout[i] = tanhf(dout[i]);
}

// ---------------------------------------------------------------------------
extern "C" void kernel_launch(void* const* d_in, const int* in_sizes, int n_in,
                              void* d_out, int out_size, void* d_ws, size_t ws_size,
                              hipStream_t stream)
{
    const float* input = (const float*)d_in[0];   // [B, IN]
    const float* wts   = (const float*)d_in[1];   // [E]
    const int*   conn  = (const int*)  d_in[2];   // [2, E]
    const int*   pN    = (const int*)  d_in[3];   // scalar on device
    const int*   pIN   = (const int*)  d_in[4];   // scalar on device
    const int*   pOUT  = (const int*)  d_in[5];   // scalar on device

    const int BIN = in_sizes[0];                  // B * IN (host-known)
    const int E   = in_sizes[1];
    const int* from = conn;
    const int* to   = conn + E;

    float* total1 = (float*)d_ws;                 // B * N floats (32 MB)
    float* dout   = (float*)d_out;                // B * OUT floats

    k_zero<<<2048, 256, 0, stream>>>(total1, pN, BIN, pIN, dout, out_size);

    const size_t lds_bytes = (size_t)BIN * sizeof(float); // 32 KB
    k_step1<<<1024, 256, lds_bytes, stream>>>(input, wts, from, to, total1,
                                              E, BIN, pN, pIN);

    k_step2<<<1024, 256, 0, stream>>>(wts, from, to, total1, dout,
                                      E, BIN, pN, pIN, pOUT);

    k_final<<<(out_size + 255) / 256, 256, 0, stream>>>(dout, out_size);
}